// NGCF_19877108646626
// MI455X (gfx1250) — compile-verified
//
#include <hip/hip_runtime.h>
#include <math.h>

#define UCNT 50000
#define NTOT 100000
#define DDIM 64
#define NNZ_CNT 3200000
#define BATCH 4096

typedef float v2f __attribute__((ext_vector_type(2)));
typedef float v8f __attribute__((ext_vector_type(8)));

// ---------------------------------------------------------------------------
// Build E0 = concat(user_embed, item_embed) into Ecur and all_e[:,0:64].
// Also zero the loss accumulators (must happen every launch).
// ---------------------------------------------------------------------------
__global__ __launch_bounds__(256) void k_concat_init(
    const float* __restrict__ ue, const float* __restrict__ ie,
    float* __restrict__ Ecur, float* __restrict__ alle,
    float* __restrict__ accum)
{
    if (blockIdx.x == 0 && threadIdx.x < 8) accum[threadIdx.x] = 0.0f;
    size_t i4 = (size_t)blockIdx.x * 256 + threadIdx.x;       // float4 index
    const size_t tot4 = (size_t)NTOT * DDIM / 4;
    if (i4 >= tot4) return;
    size_t row = i4 >> 4;                                     // 16 float4 per row
    int    c4  = (int)(i4 & 15);
    float4 v;
    if (row < UCNT) v = ((const float4*)ue)[row * 16 + c4];
    else            v = ((const float4*)ie)[(row - UCNT) * 16 + c4];
    ((float4*)Ecur)[i4] = v;
    ((float4*)(alle + row * (size_t)(4 * DDIM)))[c4] = v;     // slot 0
}

// ---------------------------------------------------------------------------
// Zero a float buffer (float4 granularity).
// ---------------------------------------------------------------------------
__global__ __launch_bounds__(256) void k_zero(float* __restrict__ p, size_t n4)
{
    size_t i = (size_t)blockIdx.x * 256 + threadIdx.x;
    if (i < n4) ((float4*)p)[i] = make_float4(0.f, 0.f, 0.f, 0.f);
}

// ---------------------------------------------------------------------------
// SpMM: L_mul[rows[e]] += vals[e] * E[cols[e]]. 16 threads per edge,
// each thread handles 4 consecutive floats (float4 gather + 4 f32 atomics).
// E (25.6 MB) and L_mul are L2-resident on MI455X (192 MB L2).
// ---------------------------------------------------------------------------
__global__ __launch_bounds__(256) void k_spmm(
    const int* __restrict__ rows, const int* __restrict__ cols,
    const float* __restrict__ vals,
    const float* __restrict__ E, float* __restrict__ Lmul)
{
    size_t gid  = (size_t)blockIdx.x * 256 + threadIdx.x;
    size_t edge = gid >> 4;
    int    sub  = (int)(gid & 15);
    if (edge >= NNZ_CNT) return;
    int   r = rows[edge];
    int   c = cols[edge];
    float v = vals[edge];
    float4 e = ((const float4*)(E + (size_t)c * DDIM))[sub];
    float* dst = Lmul + (size_t)r * DDIM + sub * 4;
    atomicAdd(dst + 0, v * e.x);
    atomicAdd(dst + 1, v * e.y);
    atomicAdd(dst + 2, v * e.z);
    atomicAdd(dst + 3, v * e.w);
}

// ---------------------------------------------------------------------------
// Fused layer kernel (WMMA f32 16x16x4):
//   out = leaky_relu( (L+E)@W1 + b1 + (L*E)@W2 + b2 , 0.2 )
//   Enext = out ; all_e[:, slot*64 : ] = out / max(||out||_row, 1e-12)
// 128 threads = 4 waves; 64 rows per block; A/H staged transposed in LDS.
// ---------------------------------------------------------------------------
__global__ __launch_bounds__(128) void k_layer(
    const float* __restrict__ E, const float* __restrict__ L,
    const float* __restrict__ W1, const float* __restrict__ b1,
    const float* __restrict__ W2, const float* __restrict__ b2,
    float* __restrict__ Enext, float* __restrict__ alle, int slot)
{
    __shared__ float sAT[DDIM * 64];   // A = L+E, stored [col][row]  (16 KB)
    __shared__ float sHT[DDIM * 64];   // H = L*E, stored [col][row]  (16 KB)
    __shared__ float sW1[DDIM * DDIM]; // row-major [k][n]            (16 KB)
    __shared__ float sW2[DDIM * DDIM]; //                             (16 KB)

    const int tid     = threadIdx.x;
    const int rowBase = blockIdx.x * 64;

    // ---- stage W1/W2 (1024 float4 each) ----
    for (int i = tid; i < DDIM * DDIM / 4; i += 128) {
        ((float4*)sW1)[i] = ((const float4*)W1)[i];
        ((float4*)sW2)[i] = ((const float4*)W2)[i];
    }
    // ---- stage A/H transposed; tail rows zero-padded ----
    for (int i = tid; i < 64 * 16; i += 128) {
        int row  = i & 63;
        int c4   = i >> 6;                 // 0..15
        int grow = rowBase + row;
        float4 e = make_float4(0.f, 0.f, 0.f, 0.f);
        float4 l = e;
        if (grow < NTOT) {
            e = ((const float4*)(E + (size_t)grow * DDIM))[c4];
            l = ((const float4*)(L + (size_t)grow * DDIM))[c4];
        }
        int c = c4 * 4;
        sAT[(c + 0) * 64 + row] = e.x + l.x;
        sAT[(c + 1) * 64 + row] = e.y + l.y;
        sAT[(c + 2) * 64 + row] = e.z + l.z;
        sAT[(c + 3) * 64 + row] = e.w + l.w;
        sHT[(c + 0) * 64 + row] = e.x * l.x;
        sHT[(c + 1) * 64 + row] = e.y * l.y;
        sHT[(c + 2) * 64 + row] = e.z * l.z;
        sHT[(c + 3) * 64 + row] = e.w * l.w;
    }
    __syncthreads();

    const int wave = tid >> 5;
    const int lane = tid & 31;
    const int hi   = lane >> 4;            // 0 or 1
    const int lo   = lane & 15;
    const int arow = wave * 16 + lo;       // A-matrix row within block tile

    // C init = b1 + b2 broadcast per output column (depends only on N = nt*16+lo)
    v8f acc[4];
#pragma unroll
    for (int nt = 0; nt < 4; ++nt) {
        float bs = b1[nt * 16 + lo] + b2[nt * 16 + lo];
        acc[nt] = (v8f){bs, bs, bs, bs, bs, bs, bs, bs};
    }

    // K = 64 in steps of 4; two chained f32 WMMAs (A@W1 then H@W2) per tile
    for (int k0 = 0; k0 < DDIM; k0 += 4) {
        int ka = k0 + 2 * hi;
        v2f a, h;
        a.x = sAT[(ka + 0) * 64 + arow];
        a.y = sAT[(ka + 1) * 64 + arow];
        h.x = sHT[(ka + 0) * 64 + arow];
        h.y = sHT[(ka + 1) * 64 + arow];
#pragma unroll
        for (int nt = 0; nt < 4; ++nt) {
            int ncol = nt * 16 + lo;
            v2f w1v, w2v;
            w1v.x = sW1[(ka + 0) * DDIM + ncol];
            w1v.y = sW1[(ka + 1) * DDIM + ncol];
            w2v.x = sW2[(ka + 0) * DDIM + ncol];
            w2v.y = sW2[(ka + 1) * DDIM + ncol];
            acc[nt] = __builtin_amdgcn_wmma_f32_16x16x4_f32(
                false, a, false, w1v, (short)0, acc[nt], false, false);
            acc[nt] = __builtin_amdgcn_wmma_f32_16x16x4_f32(
                false, h, false, w2v, (short)0, acc[nt], false, false);
        }
    }

    __syncthreads();   // all waves done reading sAT/sHT/sW*

    // leaky-ReLU; park result tile row-major in reused LDS (sAT)
    float* sOut = sAT;
#pragma unroll
    for (int nt = 0; nt < 4; ++nt) {
#pragma unroll
        for (int j = 0; j < 8; ++j) {
            float x = acc[nt][j];
            x = (x > 0.f) ? x : 0.2f * x;
            int m = j + 8 * hi;                         // C/D layout row
            sOut[(wave * 16 + m) * 64 + nt * 16 + lo] = x;
        }
    }
    __syncthreads();

    // per-row norm; write Enext (unnormalized) and all_e slot (normalized)
    if (tid < 64) {
        int grow = rowBase + tid;
        if (grow < NTOT) {
            const float* r = sOut + tid * 64;
            float ss = 0.f;
#pragma unroll
            for (int c = 0; c < 64; ++c) ss += r[c] * r[c];
            float nrm = sqrtf(ss);
            nrm = fmaxf(nrm, 1e-12f);
            float inv = 1.0f / nrm;
            float* en = Enext + (size_t)grow * DDIM;
            float* ae = alle + (size_t)grow * (4 * DDIM) + slot * DDIM;
#pragma unroll
            for (int c = 0; c < 64; ++c) {
                float x = r[c];
                en[c] = x;
                ae[c] = x * inv;
            }
        }
    }
}

// ---------------------------------------------------------------------------
// BPR loss + L2 partial sums. Thread per sample; block reduce; atomic accum.
//   accum[0] += sum softplus(-(pos - neg))
//   accum[1] += sum u^2 + sum p^2
//   accum[2] += sum n^2        (sqrt applied once at the end, per reference)
// ---------------------------------------------------------------------------
__global__ __launch_bounds__(256) void k_loss(
    const int* __restrict__ users, const int* __restrict__ pos,
    const int* __restrict__ neg, const float* __restrict__ alle,
    float* __restrict__ accum)
{
    __shared__ float s0[256], s1[256], s2[256];
    int b = blockIdx.x * 256 + threadIdx.x;
    float lbpr = 0.f, lup = 0.f, ln2 = 0.f;
    if (b < BATCH) {
        const float4* u = (const float4*)(alle + (size_t)users[b] * 256);
        const float4* p = (const float4*)(alle + (size_t)pos[b] * 256);
        const float4* n = (const float4*)(alle + (size_t)neg[b] * 256);
        float pr = 0.f, nr = 0.f, su = 0.f, sp = 0.f, sn = 0.f;
        for (int i = 0; i < 64; ++i) {
            float4 uv = u[i], pv = p[i], nv = n[i];
            pr += uv.x * pv.x + uv.y * pv.y + uv.z * pv.z + uv.w * pv.w;
            nr += uv.x * nv.x + uv.y * nv.y + uv.z * nv.z + uv.w * nv.w;
            su += uv.x * uv.x + uv.y * uv.y + uv.z * uv.z + uv.w * uv.w;
            sp += pv.x * pv.x + pv.y * pv.y + pv.z * pv.z + pv.w * pv.w;
            sn += nv.x * nv.x + nv.y * nv.y + nv.z * nv.z + nv.w * nv.w;
        }
        float d = pr - nr;
        // softplus(-d) = -log_sigmoid(d), numerically stable
        lbpr = (d > 0.f) ? log1pf(expf(-d)) : (-d + log1pf(expf(d)));
        lup  = su + sp;
        ln2  = sn;
    }
    s0[threadIdx.x] = lbpr;
    s1[threadIdx.x] = lup;
    s2[threadIdx.x] = ln2;
    __syncthreads();
    for (int off = 128; off > 0; off >>= 1) {
        if (threadIdx.x < (unsigned)off) {
            s0[threadIdx.x] += s0[threadIdx.x + off];
            s1[threadIdx.x] += s1[threadIdx.x + off];
            s2[threadIdx.x] += s2[threadIdx.x + off];
        }
        __syncthreads();
    }
    if (threadIdx.x == 0) {
        atomicAdd(accum + 0, s0[0]);
        atomicAdd(accum + 1, s1[0]);
        atomicAdd(accum + 2, s2[0]);
    }
}

__global__ void k_final(const float* __restrict__ accum, float* __restrict__ out)
{
    if (blockIdx.x == 0 && threadIdx.x == 0) {
        float bpr = accum[0] / (float)BATCH;
        float l2n = (accum[1] + sqrtf(accum[2])) * 0.5f;
        out[0] = bpr + 1e-5f * l2n / (float)BATCH;
    }
}

// ---------------------------------------------------------------------------
extern "C" void kernel_launch(void* const* d_in, const int* in_sizes, int n_in,
                              void* d_out, int out_size, void* d_ws, size_t ws_size,
                              hipStream_t stream)
{
    (void)in_sizes; (void)n_in; (void)out_size; (void)ws_size;
    const int*   users = (const int*)d_in[0];
    const int*   pos   = (const int*)d_in[1];
    const int*   neg   = (const int*)d_in[2];
    const int*   rows  = (const int*)d_in[3];
    const int*   cols  = (const int*)d_in[4];
    const float* vals  = (const float*)d_in[5];
    const float* ue    = (const float*)d_in[6];
    const float* ie    = (const float*)d_in[7];
    const float* W1    = (const float*)d_in[8];
    const float* b1    = (const float*)d_in[9];
    const float* W2    = (const float*)d_in[10];
    const float* b2    = (const float*)d_in[11];
    float* out = (float*)d_out;

    float* ws = (float*)d_ws;
    const size_t ND = (size_t)NTOT * DDIM;
    float* Ebuf0 = ws;                 // 25.6 MB
    float* Ebuf1 = Ebuf0 + ND;         // 25.6 MB
    float* Lmul  = Ebuf1 + ND;         // 25.6 MB
    float* alle  = Lmul  + ND;         // 102.4 MB
    float* accum = alle  + ND * 4;     // 32 B

    const size_t tot4 = ND / 4;
    k_concat_init<<<(unsigned)((tot4 + 255) / 256), 256, 0, stream>>>(
        ue, ie, Ebuf0, alle, accum);

    float* cur = Ebuf0;
    float* nxt = Ebuf1;
    for (int l = 0; l < 3; ++l) {
        k_zero<<<(unsigned)((tot4 + 255) / 256), 256, 0, stream>>>(Lmul, tot4);

        size_t spmm_threads = (size_t)NNZ_CNT * 16;
        k_spmm<<<(unsigned)((spmm_threads + 255) / 256), 256, 0, stream>>>(
            rows, cols, vals, cur, Lmul);

        unsigned gl = (NTOT + 63) / 64;
        k_layer<<<gl, 128, 0, stream>>>(
            cur, Lmul, W1 + (size_t)l * 4096, b1 + (size_t)l * 64,
            W2 + (size_t)l * 4096, b2 + (size_t)l * 64, nxt, alle, l + 1);

        float* t = cur; cur = nxt; nxt = t;
    }

    k_loss<<<BATCH / 256, 256, 0, stream>>>(users, pos, neg, alle, accum);
    k_final<<<1, 32, 0, stream>>>(accum, out);
}